// KaleidoscopeAdversary_1202590843535
// MI455X (gfx1250) — compile-verified
//
#include <hip/hip_runtime.h>
#include <cstdint>
#include <cstddef>

// Problem constants (match reference)
#define B_ 64
#define S_ 8
#define H_ 512
#define W_ 512
#define HW_ (H_ * W_)
#define TRANSP_ 0.3f

typedef float        v4f __attribute__((ext_vector_type(4)));
typedef unsigned int v4u __attribute__((ext_vector_type(4)));

// ---------------------------------------------------------------------------
// Pass 2 (placed first so the disasm snippet shows the CDNA5 async path):
// one 32x32 pixel tile per 256-thread block (8 waves, wave32).
//  - halo'd packed-mask tile (36 rows x 40 bytes) loaded via CDNA5
//    global_load_async_to_lds_b32 (ASYNCcnt) after LDS pre-zero (OOB = 0,
//    matching the reference's zero-padded convolution)
//  - separable 5x5 OR-dilation: horizontal pass in LDS, vertical in registers
//  - edge  = (dilated & ~mask) != 0  (bitwise per shape, then any-reduce)
//  - image/edge_vars streamed with nontemporal float4 loads/stores
// ---------------------------------------------------------------------------
__global__ __launch_bounds__(256) void composite_kernel(
    const float*  __restrict__ img,      // [B][3][H][W]
    const float*  __restrict__ cvars,    // [B][S][3]
    const float*  __restrict__ evars,    // [B][1][H][W]
    const float*  __restrict__ cinit,    // [B][S][3]
    const uint8_t* __restrict__ packed,  // [B][H][W]
    float* __restrict__ out)             // [B][3][H][W]
{
    __shared__ uint8_t mt[36 * 40];      // halo tile: rows r0-2..r0+33, cols c0-4..c0+35
    __shared__ uint8_t hd[36 * 32];      // horizontally dilated tile
    __shared__ float   scol[S_ * 3];     // per-shape colours for this image

    const int tid = threadIdx.x;
    const int c0  = blockIdx.x * 32;
    const int r0  = blockIdx.y * 32;
    const int b   = blockIdx.z;

    // Pre-zero halo tile (360 dwords) so out-of-image halo reads act as zero
    {
        uint32_t* z = (uint32_t*)mt;
        for (int i = tid; i < 360; i += 256) z[i] = 0u;
    }
    if (tid < S_ * 3) scol[tid] = cinit[b * (S_ * 3) + tid] + cvars[b * (S_ * 3) + tid];
    __syncthreads();

    // Async copy of the in-bounds portion of the halo tile into LDS.
    // 36 rows x 10 dwords = 360 dword transfers; tile edges are 4B aligned
    // because c0 % 32 == 0. GVS mode: copies a dword from global address saddr + vaddr to the LDS address in vdst;
    // tracked by ASYNCcnt, waited with s_wait_asynccnt.
    const uint8_t* pbase = packed + (size_t)b * HW_;
    for (int idx = tid; idx < 360; idx += 256) {
        int i  = idx / 10;
        int j4 = (idx % 10) * 4;
        int gr = r0 - 2 + i;
        int gc = c0 - 4 + j4;
        if (gr >= 0 && gr < H_ && gc >= 0 && gc < W_) {
            unsigned lds_off = (unsigned)(uintptr_t)(void*)(mt + i * 40 + j4);
            int voff = gr * W_ + gc;
            asm volatile("global_load_async_to_lds_b32 %0, %1, %2"
                         :: "v"(lds_off), "v"(voff), "s"(pbase)
                         : "memory");
        }
    }
    asm volatile("s_wait_asynccnt 0" ::: "memory");
    __syncthreads();

    // Horizontal OR-dilation: hd[i][c] = OR of mt[i][c+2 .. c+6]
    // (output col c maps to LDS col j = c+4). 288 dwords of output.
    uint32_t* hd32 = (uint32_t*)hd;
    for (int idx = tid; idx < 36 * 8; idx += 256) {
        int i  = idx >> 3;
        int cq = (idx & 7) * 4;
        const uint8_t* row = mt + i * 40;
        uint32_t r = 0;
#pragma unroll
        for (int k = 0; k < 4; ++k) {
            uint8_t v = (uint8_t)(row[cq + k + 2] | row[cq + k + 3] |
                                  row[cq + k + 4] | row[cq + k + 5] |
                                  row[cq + k + 6]);
            r |= (uint32_t)v << (8 * k);
        }
        hd32[idx] = r;
    }
    __syncthreads();

    // Hoist the 8 shape colours into registers once (loop-invariant).
    float col_r[S_], col_g[S_], col_b[S_];
#pragma unroll
    for (int s = 0; s < S_; ++s) {
        col_r[s] = scol[s * 3 + 0];
        col_g[s] = scol[s * 3 + 1];
        col_b[s] = scol[s * 3 + 2];
    }

    // Each thread finishes 4 consecutive pixels of one row (float4 lane).
    const int r  = tid >> 3;          // 0..31
    const int cq = (tid & 7) * 4;     // 0,4,...,28

    // Vertical OR-dilation over 5 rows (pixel row r -> hd rows r..r+4)
    uint32_t dil = hd32[(r + 0) * 8 + (cq >> 2)];
    dil |= hd32[(r + 1) * 8 + (cq >> 2)];
    dil |= hd32[(r + 2) * 8 + (cq >> 2)];
    dil |= hd32[(r + 3) * 8 + (cq >> 2)];
    dil |= hd32[(r + 4) * 8 + (cq >> 2)];
    const uint32_t cen = *(const uint32_t*)(mt + (r + 2) * 40 + cq + 4);
    const uint32_t edg = dil & ~cen;  // per-shape edge bits, per pixel byte

    const size_t p     = (size_t)(r0 + r) * W_ + (size_t)(c0 + cq);
    const size_t plane = (size_t)b * 3 * HW_;
    const v4f i0 = __builtin_nontemporal_load((const v4f*)(img + plane + 0 * HW_ + p));
    const v4f i1 = __builtin_nontemporal_load((const v4f*)(img + plane + 1 * HW_ + p));
    const v4f i2 = __builtin_nontemporal_load((const v4f*)(img + plane + 2 * HW_ + p));
    const v4f ev = __builtin_nontemporal_load((const v4f*)(evars + (size_t)b * HW_ + p));

    v4f o0, o1, o2;
#pragma unroll
    for (int k = 0; k < 4; ++k) {
        const uint32_t mb = (cen >> (8 * k)) & 0xFFu;
        const float a = (mb != 0u) ? TRANSP_ : 1.0f;   // 1 - combined*(1-T)
        float s0 = 0.f, s1 = 0.f, s2 = 0.f;
#pragma unroll
        for (int s = 0; s < S_; ++s) {
            const float f = (float)((mb >> s) & 1u);
            s0 += f * col_r[s];
            s1 += f * col_g[s];
            s2 += f * col_b[s];
        }
        float v0 = a * i0[k] + s0;
        float v1 = a * i1[k] + s1;
        float v2 = a * i2[k] + s2;
        if ((edg >> (8 * k)) & 0xFFu) {
            const float e = __builtin_fabsf(ev[k]);
            v0 = e; v1 = e; v2 = e;
        }
        o0[k] = fminf(fmaxf(v0, 0.f), 1.f);
        o1[k] = fminf(fmaxf(v1, 0.f), 1.f);
        o2[k] = fminf(fmaxf(v2, 0.f), 1.f);
    }

    __builtin_nontemporal_store(o0, (v4f*)(out + plane + 0 * HW_ + p));
    __builtin_nontemporal_store(o1, (v4f*)(out + plane + 1 * HW_ + p));
    __builtin_nontemporal_store(o2, (v4f*)(out + plane + 2 * HW_ + p));
}

// ---------------------------------------------------------------------------
// Pass 1: pack 8 boolean masks (1 byte each, values 0/1) into one byte/pixel.
// SWAR: since every mask byte is 0 or 1, (dword << s) keeps bit s inside its
// own byte, so OR-accumulating builds the 8-bit shape bitmap per pixel.
// Each thread packs 16 pixels with 128-bit loads/stores. The bitmap store is
// deliberately regular-temporal: 16 MiB fits in the 192 MB L2 and is re-read
// by pass 2.
// ---------------------------------------------------------------------------
__global__ __launch_bounds__(256) void pack_masks_kernel(
    const uint8_t* __restrict__ masks,   // [B][S][H][W] bool bytes
    uint8_t* __restrict__ packed)        // [B][H][W] bitmap
{
    size_t t  = (size_t)blockIdx.x * 256 + threadIdx.x;
    size_t p0 = t * 16;                  // pixel index in [0, B*HW)
    size_t b  = p0 / HW_;
    size_t q  = p0 % HW_;                // HW_ % 16 == 0 -> never straddles images
    const v4u* base = (const v4u*)(masks + b * (size_t)S_ * HW_ + q);
    v4u acc = {0u, 0u, 0u, 0u};
#pragma unroll
    for (int s = 0; s < S_; ++s) {
        v4u m = __builtin_nontemporal_load(base + (size_t)s * (HW_ / 16));
        acc |= m << (unsigned)s;
    }
    *(v4u*)(packed + p0) = acc;
}

// ---------------------------------------------------------------------------
extern "C" void kernel_launch(void* const* d_in, const int* in_sizes, int n_in,
                              void* d_out, int out_size, void* d_ws, size_t ws_size,
                              hipStream_t stream) {
    const float*   img   = (const float*)d_in[0];     // image_batch [B,3,H,W] f32
    const float*   cvars = (const float*)d_in[1];     // colour_vars [B,S,3]  f32
    const float*   evars = (const float*)d_in[2];     // edge_vars   [B,1,H,W] f32
    const float*   cinit = (const float*)d_in[3];     // colour_init [B,S,3]  f32
    const uint8_t* masks = (const uint8_t*)d_in[4];   // shape_masks [B,S,H,W] bool
    float*         outp  = (float*)d_out;             // [B,3,H,W] f32
    uint8_t*       packd = (uint8_t*)d_ws;            // B*HW = 16 MiB scratch

    // Pass 1: 16.7M pixels / 16 per thread / 256 per block = 4096 blocks
    pack_masks_kernel<<<dim3((B_ * HW_) / 16 / 256), dim3(256), 0, stream>>>(masks, packd);

    // Pass 2: one 32x32 tile per block
    composite_kernel<<<dim3(W_ / 32, H_ / 32, B_), dim3(256), 0, stream>>>(
        img, cvars, evars, cinit, packd, outp);
}